// ScatLayerj1_3135326126128
// MI455X (gfx1250) — compile-verified
//
#include <hip/hip_runtime.h>
#include <math.h>

#define NB 32
#define CB 64
#define H 128
#define W 128
#define HO 64
#define WO 64
#define PLANE (H * W)     // 16384
#define OPLANE (HO * WO)  // 4096
#define NTHREADS 1024

#if __has_builtin(__builtin_amdgcn_global_load_async_to_lds_b128) && \
    __has_builtin(__builtin_amdgcn_s_wait_asynccnt)
#define USE_ASYNC_LDS 1
#else
#define USE_ASYNC_LDS 0
#endif

typedef int v4i __attribute__((ext_vector_type(4)));
typedef __attribute__((address_space(1))) v4i GV4;  // global int4
typedef __attribute__((address_space(3))) v4i LV4;  // LDS int4

__device__ __forceinline__ int symm128(int i) {
    // half-sample symmetric reflection for n=128, valid for i in [-128, 255]
    if (i < 0) i = -1 - i;
    if (i >= 128) i = 255 - i;
    return i;
}

__global__ __launch_bounds__(NTHREADS) void scat_j1_kernel(
    const float* __restrict__ x,
    const float* __restrict__ h0o,
    const float* __restrict__ h1o,
    float* __restrict__ out)
{
    __shared__ float xs[PLANE];   // 64 KB
    __shared__ float slo[PLANE];  // 64 KB
    __shared__ float shi[PLANE];  // 64 KB  (192 KB total, <= 320 KB/WGP)

    const int p   = blockIdx.x;       // plane = n*CB + c
    const int n   = p / CB;
    const int c   = p % CB;
    const int tid = threadIdx.x;

    const float* xg = x + (size_t)p * PLANE;

    // Filter taps (uniform, L2-resident)
    float f0[5], f1[7];
#pragma unroll
    for (int k = 0; k < 5; ++k) f0[k] = h0o[k];
#pragma unroll
    for (int k = 0; k < 7; ++k) f1[k] = h1o[k];

    // ---------- Phase A: stage full plane into LDS (async b128 copies) ----------
#pragma unroll
    for (int k = 0; k < PLANE / 4 / NTHREADS; ++k) {  // 4 iters, 16B per lane
        int idx = tid + k * NTHREADS;                  // float4 index
#if USE_ASYNC_LDS
        __builtin_amdgcn_global_load_async_to_lds_b128(
            (GV4*)(xg + 4 * idx), (LV4*)(xs + 4 * idx), 0, 0);
#else
        ((float4*)xs)[idx] = ((const float4*)(xg))[idx];
#endif
    }
#if USE_ASYNC_LDS
    __builtin_amdgcn_s_wait_asynccnt(0);
#endif
    __syncthreads();

    // ---------- Phase B: horizontal (W-axis) filters -> slo, shi ----------
#pragma unroll
    for (int k = 0; k < PLANE / NTHREADS; ++k) {  // 16 iters
        int q  = tid + k * NTHREADS;
        int y  = q >> 7;
        int xc = q & 127;
        const float* row = xs + (y << 7);
        float lo, hi;
        if (xc >= 3 && xc < 125) {
            float a0 = row[xc - 3], a1 = row[xc - 2], a2 = row[xc - 1],
                  a3 = row[xc],     a4 = row[xc + 1], a5 = row[xc + 2],
                  a6 = row[xc + 3];
            lo = f0[0]*a1 + f0[1]*a2 + f0[2]*a3 + f0[3]*a4 + f0[4]*a5;
            hi = f1[0]*a0 + f1[1]*a1 + f1[2]*a2 + f1[3]*a3 + f1[4]*a4
               + f1[5]*a5 + f1[6]*a6;
        } else {
            float a[7];
#pragma unroll
            for (int t = 0; t < 7; ++t) a[t] = row[symm128(xc - 3 + t)];
            lo = f0[0]*a[1] + f0[1]*a[2] + f0[2]*a[3] + f0[3]*a[4] + f0[4]*a[5];
            hi = f1[0]*a[0] + f1[1]*a[1] + f1[2]*a[2] + f1[3]*a[3] + f1[4]*a[4]
               + f1[5]*a[5] + f1[6]*a[6];
        }
        slo[q] = lo;
        shi[q] = hi;
    }
    __syncthreads();

    // ---------- Phase C: vertical filters + q2c + magnitude + avg-pool ----------
    const float scale   = 0.70710678118654752f;  // 1/sqrt(2)
    const float magbias = 0.01f;

    float* ll_out = out + (size_t)p * OPLANE;
    float* r_base = out + (size_t)NB * CB * OPLANE
                  + (size_t)n * 6 * CB * OPLANE + (size_t)c * OPLANE;
    const size_t sstride = (size_t)CB * OPLANE;  // subband stride in r

#pragma unroll
    for (int k = 0; k < OPLANE / NTHREADS; ++k) {  // 4 iters
        int q  = tid + k * NTHREADS;
        int i  = q >> 6;
        int j  = q & 63;
        int y0 = 2 * i;
        int x0 = 2 * j;

        int r[8];
#pragma unroll
        for (int t = 0; t < 8; ++t) r[t] = symm128(y0 - 3 + t) << 7;

        float L0[8], L1[8], G0[8], G1[8];
#pragma unroll
        for (int t = 0; t < 8; ++t) {
            L0[t] = slo[r[t] + x0];
            L1[t] = slo[r[t] + x0 + 1];
            G0[t] = shi[r[t] + x0];
            G1[t] = shi[r[t] + x0 + 1];
        }

        // 7-tap vertical (h1): lh (from lo), hh (from hi), at both row parities
        float lh_e0 = 0, lh_e1 = 0, lh_o0 = 0, lh_o1 = 0;
        float hh_e0 = 0, hh_e1 = 0, hh_o0 = 0, hh_o1 = 0;
#pragma unroll
        for (int t = 0; t < 7; ++t) {
            lh_e0 += f1[t] * L0[t];     lh_e1 += f1[t] * L1[t];
            lh_o0 += f1[t] * L0[t + 1]; lh_o1 += f1[t] * L1[t + 1];
            hh_e0 += f1[t] * G0[t];     hh_e1 += f1[t] * G1[t];
            hh_o0 += f1[t] * G0[t + 1]; hh_o1 += f1[t] * G1[t + 1];
        }
        // 5-tap vertical (h0): ll (from lo), hl (from hi)
        float ll_e0 = 0, ll_e1 = 0, ll_o0 = 0, ll_o1 = 0;
        float hl_e0 = 0, hl_e1 = 0, hl_o0 = 0, hl_o1 = 0;
#pragma unroll
        for (int t = 0; t < 5; ++t) {
            ll_e0 += f0[t] * L0[t + 1]; ll_e1 += f0[t] * L1[t + 1];
            ll_o0 += f0[t] * L0[t + 2]; ll_o1 += f0[t] * L1[t + 2];
            hl_e0 += f0[t] * G0[t + 1]; hl_e1 += f0[t] * G1[t + 1];
            hl_o0 += f0[t] * G0[t + 2]; hl_o1 += f0[t] * G1[t + 2];
        }

        // ll: 2x2 average pool
        ll_out[q] = 0.25f * (ll_e0 + ll_e1 + ll_o0 + ll_o1);

        // q2c: a=y[2i,2j], b=y[2i,2j+1], c=y[2i+1,2j], d=y[2i+1,2j+1] (x 1/sqrt2)
        // first=(a-d, b+c) ; second=(a+d, b-c)
        float b2 = magbias * magbias;
        {   // lh -> d15 (s0), d165 (s5)
            float a = scale * lh_e0, b = scale * lh_e1,
                  cc = scale * lh_o0, d = scale * lh_o1;
            float re0 = a - d, im0 = b + cc, re1 = a + d, im1 = b - cc;
            r_base[0 * sstride + q] = sqrtf(re0 * re0 + im0 * im0 + b2) - magbias;
            r_base[5 * sstride + q] = sqrtf(re1 * re1 + im1 * im1 + b2) - magbias;
        }
        {   // hh -> d45 (s1), d135 (s4)
            float a = scale * hh_e0, b = scale * hh_e1,
                  cc = scale * hh_o0, d = scale * hh_o1;
            float re0 = a - d, im0 = b + cc, re1 = a + d, im1 = b - cc;
            r_base[1 * sstride + q] = sqrtf(re0 * re0 + im0 * im0 + b2) - magbias;
            r_base[4 * sstride + q] = sqrtf(re1 * re1 + im1 * im1 + b2) - magbias;
        }
        {   // hl -> d75 (s2), d105 (s3)
            float a = scale * hl_e0, b = scale * hl_e1,
                  cc = scale * hl_o0, d = scale * hl_o1;
            float re0 = a - d, im0 = b + cc, re1 = a + d, im1 = b - cc;
            r_base[2 * sstride + q] = sqrtf(re0 * re0 + im0 * im0 + b2) - magbias;
            r_base[3 * sstride + q] = sqrtf(re1 * re1 + im1 * im1 + b2) - magbias;
        }
    }
}

extern "C" void kernel_launch(void* const* d_in, const int* in_sizes, int n_in,
                              void* d_out, int out_size, void* d_ws, size_t ws_size,
                              hipStream_t stream) {
    (void)in_sizes; (void)n_in; (void)out_size; (void)d_ws; (void)ws_size;
    const float* x   = (const float*)d_in[0];
    const float* h0o = (const float*)d_in[1];
    const float* h1o = (const float*)d_in[2];
    float* out = (float*)d_out;
    scat_j1_kernel<<<dim3(NB * CB), dim3(NTHREADS), 0, stream>>>(x, h0o, h1o, out);
}